// ConnectionG2C_22299470201305
// MI455X (gfx1250) — compile-verified
//
#include <hip/hip_runtime.h>
#include <math.h>

typedef __attribute__((ext_vector_type(2))) float v2f;
typedef __attribute__((ext_vector_type(8))) float v8f;

#define BD 256
#define PCHUNK 16

__device__ __forceinline__ v8f wmma4(v2f a, v2f b, v8f c) {
  // V_WMMA_F32_16X16X4_F32 : D = A(16x4) * B(4x16) + C(16x16), all fp32
  return __builtin_amdgcn_wmma_f32_16x16x4_f32(
      /*neg_a=*/false, a, /*neg_b=*/false, b,
      /*c_mod=*/(short)0, c, /*reuse_a=*/false, /*reuse_b=*/false);
}

// ---------------------------------------------------------------------------
// K/V projection: K[b,o,n] = sum_g Wk[o,g]*graph[b,n,g] + bk[o]  (G=32, tiny)
// ---------------------------------------------------------------------------
__global__ void kv_proj(const float* __restrict__ graph,
                        const float* __restrict__ Wk, const float* __restrict__ bk,
                        const float* __restrict__ Wv, const float* __restrict__ bv,
                        float* __restrict__ K, float* __restrict__ V,
                        int Bn, int C, int N, int G) {
  long idx = (long)blockIdx.x * blockDim.x + threadIdx.x;
  long total = (long)Bn * C * N;
  if (idx >= total) return;
  int n = (int)(idx % N);
  long t = idx / N;
  int o = (int)(t % C);
  int b = (int)(t / C);
  const float* g  = graph + ((long)b * N + n) * G;
  const float* wk = Wk + (long)o * G;
  const float* wv = Wv + (long)o * G;
  float sk = 0.f, sv = 0.f;
  for (int i = 0; i < G; ++i) { float gv = g[i]; sk += wk[i] * gv; sv += wv[i] * gv; }
  K[idx] = sk + bk[o];
  V[idx] = sv + bv[o];
}

// ---------------------------------------------------------------------------
// Generic WMMA GEMM: Out[b,o,p] = sum_c W[o,c] * X[b,c,p]  (+ epilogue)
// 32(M) x 64(N) supertile per wave: 2 A fragments, 8 accumulators,
// 8 WMMAs per K-step against one set of B loads (4x A-reuse, 2x B-reuse).
// EP 0: Q proj   -> (acc + e0[o]) * (1/16)
// EP 1: BN+leaky
// EP 3: residual -> res + acc + e0[o]
// ---------------------------------------------------------------------------
template <int EP>
__global__ void gemm_wx(const float* __restrict__ Wm, const float* __restrict__ X,
                        float* __restrict__ Out,
                        const float* __restrict__ e0, const float* __restrict__ e1,
                        const float* __restrict__ e2, const float* __restrict__ e3,
                        const float* __restrict__ res,
                        int Bn, int O, int C, int P) {
  const int lane = threadIdx.x & 31;
  const int lo = lane & 15, hi = lane >> 4;
  const int wpb = blockDim.x >> 5;
  long tile = (long)blockIdx.x * wpb + (threadIdx.x >> 5);
  const int mT = O >> 5, nT = P >> 6;
  const long perB = (long)mT * nT;
  if (tile >= (long)Bn * perB) return;
  const int b = (int)(tile / perB);
  const long r = tile % perB;
  const int m0 = (int)(r / nT) << 5;
  const int p0 = (int)(r % nT) << 6;

  const float* A0 = Wm + (long)(m0 + lo) * C + 2 * hi;
  const float* A1 = A0 + (long)16 * C;
  const float* Bb = X + ((long)b * C + 2 * hi) * P + p0 + lo;

  v8f c00 = {}, c01 = {}, c02 = {}, c03 = {};
  v8f c10 = {}, c11 = {}, c12 = {}, c13 = {};
  for (int k = 0; k < C; k += 4) {
    v2f av0 = *(const v2f*)(A0 + k);
    v2f av1 = *(const v2f*)(A1 + k);
    const float* bp = Bb + (long)k * P;
    v2f b0v = { bp[0],  bp[P]      };
    v2f b1v = { bp[16], bp[P + 16] };
    v2f b2v = { bp[32], bp[P + 32] };
    v2f b3v = { bp[48], bp[P + 48] };
    c00 = wmma4(av0, b0v, c00); c01 = wmma4(av0, b1v, c01);
    c02 = wmma4(av0, b2v, c02); c03 = wmma4(av0, b3v, c03);
    c10 = wmma4(av1, b0v, c10); c11 = wmma4(av1, b1v, c11);
    c12 = wmma4(av1, b2v, c12); c13 = wmma4(av1, b3v, c13);
  }

  float* Ob = Out + (long)b * O * P;
#pragma unroll
  for (int half = 0; half < 2; ++half) {
    v8f q0v = half ? c10 : c00, q1v = half ? c11 : c01;
    v8f q2v = half ? c12 : c02, q3v = half ? c13 : c03;
#pragma unroll
    for (int rr = 0; rr < 8; ++rr) {
      int m = m0 + 16 * half + rr + 8 * hi;
      float q0 = q0v[rr], q1 = q1v[rr], q2 = q2v[rr], q3 = q3v[rr];
      if (EP == 0) {
        float bq = e0[m];
        q0 = (q0 + bq) * 0.0625f; q1 = (q1 + bq) * 0.0625f;
        q2 = (q2 + bq) * 0.0625f; q3 = (q3 + bq) * 0.0625f;
      } else if (EP == 1) {
        float sc = e0[m] * rsqrtf(e3[m] + 1e-5f);
        float mu = e2[m], be = e1[m];
        q0 = (q0 - mu) * sc + be; q1 = (q1 - mu) * sc + be;
        q2 = (q2 - mu) * sc + be; q3 = (q3 - mu) * sc + be;
        q0 = q0 > 0.f ? q0 : 0.1f * q0; q1 = q1 > 0.f ? q1 : 0.1f * q1;
        q2 = q2 > 0.f ? q2 : 0.1f * q2; q3 = q3 > 0.f ? q3 : 0.1f * q3;
      } else if (EP == 3) {
        float bb = e0[m];
        const float* rp = res + ((long)b * O + m) * P + p0 + lo;
        q0 += bb + rp[0]; q1 += bb + rp[16]; q2 += bb + rp[32]; q3 += bb + rp[48];
      }
      float* op = Ob + (long)m * P + p0 + lo;
      op[0] = q0; op[16] = q1; op[32] = q2; op[48] = q3;
    }
  }
}

// ---------------------------------------------------------------------------
// Attention logits: S[b,p,n] = sum_c Q[b,c,p] * K[b,c,n]   (Q pre-scaled 1/16)
// 32(M=p) x 64(N=n) supertile per wave.
// ---------------------------------------------------------------------------
__global__ void att_gemm(const float* __restrict__ Q, const float* __restrict__ K,
                         float* __restrict__ S, int Bn, int C, int P, int N) {
  const int lane = threadIdx.x & 31;
  const int lo = lane & 15, hi = lane >> 4;
  const int wpb = blockDim.x >> 5;
  long tile = (long)blockIdx.x * wpb + (threadIdx.x >> 5);
  const int mT = P >> 5, nT = N >> 6;
  const long perB = (long)mT * nT;
  if (tile >= (long)Bn * perB) return;
  const int b = (int)(tile / perB);
  const long r = tile % perB;
  const int m0 = (int)(r / nT) << 5;
  const int n0 = (int)(r % nT) << 6;

  const float* Qb = Q + (long)b * C * P + (long)(2 * hi) * P + (m0 + lo);
  const float* Kb = K + (long)b * C * N + (long)(2 * hi) * N + (n0 + lo);

  v8f c00 = {}, c01 = {}, c02 = {}, c03 = {};
  v8f c10 = {}, c11 = {}, c12 = {}, c13 = {};
  for (int k = 0; k < C; k += 4) {
    const float* qp = Qb + (long)k * P;
    v2f av0 = { qp[0],  qp[P]      };
    v2f av1 = { qp[16], qp[P + 16] };
    const float* kp = Kb + (long)k * N;
    v2f b0v = { kp[0],  kp[N]      };
    v2f b1v = { kp[16], kp[N + 16] };
    v2f b2v = { kp[32], kp[N + 32] };
    v2f b3v = { kp[48], kp[N + 48] };
    c00 = wmma4(av0, b0v, c00); c01 = wmma4(av0, b1v, c01);
    c02 = wmma4(av0, b2v, c02); c03 = wmma4(av0, b3v, c03);
    c10 = wmma4(av1, b0v, c10); c11 = wmma4(av1, b1v, c11);
    c12 = wmma4(av1, b2v, c12); c13 = wmma4(av1, b3v, c13);
  }
  float* Sb = S + (long)b * P * N;
#pragma unroll
  for (int half = 0; half < 2; ++half) {
    v8f q0v = half ? c10 : c00, q1v = half ? c11 : c01;
    v8f q2v = half ? c12 : c02, q3v = half ? c13 : c03;
#pragma unroll
    for (int rr = 0; rr < 8; ++rr) {
      int m = m0 + 16 * half + rr + 8 * hi;
      float* sp = Sb + (long)m * N + n0 + lo;
      sp[0] = q0v[rr]; sp[16] = q1v[rr]; sp[32] = q2v[rr]; sp[48] = q3v[rr];
    }
  }
}

// ---------------------------------------------------------------------------
// Column softmax over P axis, chunk-parallel
// ---------------------------------------------------------------------------
__global__ void sm_pmax(const float* __restrict__ S, float* __restrict__ pm,
                        int Bn, int P, int N) {
  int nb256 = N >> 8;
  int chunk = blockIdx.x % PCHUNK;
  int nb = (blockIdx.x / PCHUNK) % nb256;
  int b  = blockIdx.x / (PCHUNK * nb256);
  int n = (nb << 8) + threadIdx.x;
  int rows = P / PCHUNK;
  const float* p = S + ((long)b * P + (long)chunk * rows) * N + n;
  float m = -3.402823466e38f;
  for (int i = 0; i < rows; ++i) m = fmaxf(m, p[(long)i * N]);
  pm[(long)chunk * Bn * N + (long)b * N + n] = m;
}

__global__ void sm_rmax(const float* __restrict__ pm, float* __restrict__ cm, int BN) {
  int i = blockIdx.x * blockDim.x + threadIdx.x;
  if (i >= BN) return;
  float m = -3.402823466e38f;
  for (int c = 0; c < PCHUNK; ++c) m = fmaxf(m, pm[(long)c * BN + i]);
  cm[i] = m;
}

__global__ void sm_expsum(float* __restrict__ S, const float* __restrict__ cm,
                          float* __restrict__ ps, int Bn, int P, int N) {
  int nb256 = N >> 8;
  int chunk = blockIdx.x % PCHUNK;
  int nb = (blockIdx.x / PCHUNK) % nb256;
  int b  = blockIdx.x / (PCHUNK * nb256);
  int n = (nb << 8) + threadIdx.x;
  int rows = P / PCHUNK;
  float mx = cm[(long)b * N + n];
  float* p = S + ((long)b * P + (long)chunk * rows) * N + n;
  float s = 0.f;
  for (int i = 0; i < rows; ++i) {
    float v = expf(p[(long)i * N] - mx);
    p[(long)i * N] = v;
    s += v;
  }
  ps[(long)chunk * Bn * N + (long)b * N + n] = s;
}

__global__ void sm_rsum(const float* __restrict__ ps, float* __restrict__ cs, int BN) {
  int i = blockIdx.x * blockDim.x + threadIdx.x;
  if (i >= BN) return;
  float s = 0.f;
  for (int c = 0; c < PCHUNK; ++c) s += ps[(long)c * BN + i];
  cs[i] = s;
}

// Fold 1/colsum[n] into V so the message pass is a plain GEMM.
__global__ void v_scale(float* __restrict__ V, const float* __restrict__ cs,
                        int Bn, int C, int N) {
  long i = (long)blockIdx.x * blockDim.x + threadIdx.x;
  long total = (long)Bn * C * N;
  if (i >= total) return;
  int n = (int)(i % N);
  int b = (int)(i / ((long)C * N));
  V[i] *= 1.0f / cs[(long)b * N + n];
}

// ---------------------------------------------------------------------------
// Message: msg[b,c,p] = sum_n Vs[b,c,n] * E[b,p,n]
// 32(M=c) x 64(N=p) supertile; both A and B are contiguous b64 loads.
// ---------------------------------------------------------------------------
__global__ void msg_gemm(const float* __restrict__ Vs, const float* __restrict__ E,
                         float* __restrict__ Out, int Bn, int C, int N, int P) {
  const int lane = threadIdx.x & 31;
  const int lo = lane & 15, hi = lane >> 4;
  const int wpb = blockDim.x >> 5;
  long tile = (long)blockIdx.x * wpb + (threadIdx.x >> 5);
  const int mT = C >> 5, nT = P >> 6;
  const long perB = (long)mT * nT;
  if (tile >= (long)Bn * perB) return;
  const int b = (int)(tile / perB);
  const long r = tile % perB;
  const int m0 = (int)(r / nT) << 5;
  const int p0 = (int)(r % nT) << 6;

  const float* A0 = Vs + ((long)b * C + m0 + lo) * N + 2 * hi;
  const float* A1 = A0 + (long)16 * N;
  const float* Eb = E + ((long)b * P + p0 + lo) * N + 2 * hi;
  const long j16 = (long)16 * N;

  v8f c00 = {}, c01 = {}, c02 = {}, c03 = {};
  v8f c10 = {}, c11 = {}, c12 = {}, c13 = {};
  for (int k = 0; k < N; k += 4) {
    v2f av0 = *(const v2f*)(A0 + k);
    v2f av1 = *(const v2f*)(A1 + k);
    v2f b0v = *(const v2f*)(Eb + k);
    v2f b1v = *(const v2f*)(Eb + j16 + k);
    v2f b2v = *(const v2f*)(Eb + 2 * j16 + k);
    v2f b3v = *(const v2f*)(Eb + 3 * j16 + k);
    c00 = wmma4(av0, b0v, c00); c01 = wmma4(av0, b1v, c01);
    c02 = wmma4(av0, b2v, c02); c03 = wmma4(av0, b3v, c03);
    c10 = wmma4(av1, b0v, c10); c11 = wmma4(av1, b1v, c11);
    c12 = wmma4(av1, b2v, c12); c13 = wmma4(av1, b3v, c13);
  }
  float* Ob = Out + (long)b * C * P;
#pragma unroll
  for (int half = 0; half < 2; ++half) {
    v8f q0v = half ? c10 : c00, q1v = half ? c11 : c01;
    v8f q2v = half ? c12 : c02, q3v = half ? c13 : c03;
#pragma unroll
    for (int rr = 0; rr < 8; ++rr) {
      int m = m0 + 16 * half + rr + 8 * hi;
      float* op = Ob + (long)m * P + p0 + lo;
      op[0] = q0v[rr]; op[16] = q1v[rr]; op[32] = q2v[rr]; op[48] = q3v[rr];
    }
  }
}

// Transpose 3x3 weights [O][C][3][3] -> [9][O][C] for contiguous b64 A loads.
__global__ void transpose_w9(const float* __restrict__ w, float* __restrict__ wt,
                             int O, int C) {
  long idx = (long)blockIdx.x * blockDim.x + threadIdx.x;
  long total = (long)O * C * 9;
  if (idx >= total) return;
  int t9 = (int)(idx % 9);
  long oc = idx / 9;
  int c = (int)(oc % C);
  int o = (int)(oc / C);
  wt[(long)t9 * O * C + (long)o * C + c] = w[idx];
}

// ---------------------------------------------------------------------------
// 3x3 conv as 9 shifted WMMA GEMMs, 32(M) x 64(N) tile (one image row, W=64).
// Row validity uniform; column edges via clamp + 0/1 mask (EXEC stays all-1s).
// ---------------------------------------------------------------------------
__global__ void conv3x3_wmma(const float* __restrict__ wt,   // [9][O][C]
                             const float* __restrict__ Hin,  // [B][C][P]
                             const float* __restrict__ bias, // [O]
                             float* __restrict__ Out,        // [B][O][P]
                             int Bn, int O, int C, int Hd) {
  const int Wd = 64;
  const int P = Wd * Hd;
  const int lane = threadIdx.x & 31;
  const int lo = lane & 15, hi = lane >> 4;
  const int wpb = blockDim.x >> 5;
  long tile = (long)blockIdx.x * wpb + (threadIdx.x >> 5);
  const int mT = O >> 5, nT = P >> 6;
  const long perB = (long)mT * nT;
  if (tile >= (long)Bn * perB) return;
  const int b = (int)(tile / perB);
  const long r = tile % perB;
  const int m0 = (int)(r / nT) << 5;
  const int nt = (int)(r % nT);
  const int y = nt;
  const int p0 = nt << 6;

  v8f c00 = {}, c01 = {}, c02 = {}, c03 = {};
  v8f c10 = {}, c11 = {}, c12 = {}, c13 = {};
  for (int dy = 0; dy < 3; ++dy) {
    int yy = y + dy - 1;
    if (yy < 0 || yy >= Hd) continue;  // uniform branch
    for (int dx = 0; dx < 3; ++dx) {
      int t9 = dy * 3 + dx;
      int x0 = lo + dx - 1,      x1 = lo + 16 + dx - 1;
      int x2 = lo + 32 + dx - 1, x3 = lo + 48 + dx - 1;
      float k0 = (x0 >= 0 && x0 < Wd) ? 1.f : 0.f;
      float k1 = (x1 >= 0 && x1 < Wd) ? 1.f : 0.f;
      float k2 = (x2 >= 0 && x2 < Wd) ? 1.f : 0.f;
      float k3 = (x3 >= 0 && x3 < Wd) ? 1.f : 0.f;
      int o0 = min(max(x0, 0), Wd - 1), o1 = min(max(x1, 0), Wd - 1);
      int o2 = min(max(x2, 0), Wd - 1), o3 = min(max(x3, 0), Wd - 1);
      const float* A0 = wt + (long)t9 * O * C + (long)(m0 + lo) * C + 2 * hi;
      const float* A1 = A0 + (long)16 * C;
      const float* Bb = Hin + ((long)b * C + 2 * hi) * P + (long)yy * Wd;
      for (int c = 0; c < C; c += 4) {
        v2f av0 = *(const v2f*)(A0 + c);
        v2f av1 = *(const v2f*)(A1 + c);
        const float* bp = Bb + (long)c * P;
        v2f b0v = { bp[o0] * k0, bp[P + o0] * k0 };
        v2f b1v = { bp[o1] * k1, bp[P + o1] * k1 };
        v2f b2v = { bp[o2] * k2, bp[P + o2] * k2 };
        v2f b3v = { bp[o3] * k3, bp[P + o3] * k3 };
        c00 = wmma4(av0, b0v, c00); c01 = wmma4(av0, b1v, c01);
        c02 = wmma4(av0, b2v, c02); c03 = wmma4(av0, b3v, c03);
        c10 = wmma4(av1, b0v, c10); c11 = wmma4(av1, b1v, c11);
        c12 = wmma4(av1, b2v, c12); c13 = wmma4(av1, b3v, c13);
      }
    }
  }
  float* Ob = Out + (long)b * O * P;
#pragma unroll
  for (int half = 0; half < 2; ++half) {
    v8f q0v = half ? c10 : c00, q1v = half ? c11 : c01;
    v8f q2v = half ? c12 : c02, q3v = half ? c13 : c03;
#pragma unroll
    for (int rr = 0; rr < 8; ++rr) {
      int m = m0 + 16 * half + rr + 8 * hi;
      float bb = bias[m];
      float* op = Ob + (long)m * P + p0 + lo;
      op[0]  = q0v[rr] + bb; op[16] = q1v[rr] + bb;
      op[32] = q2v[rr] + bb; op[48] = q3v[rr] + bb;
    }
  }
}

// ---------------------------------------------------------------------------
extern "C" void kernel_launch(void* const* d_in, const int* in_sizes, int n_in,
                              void* d_out, int out_size, void* d_ws, size_t ws_size,
                              hipStream_t stream) {
  (void)in_sizes; (void)n_in; (void)out_size; (void)ws_size;
  const int B = 8, C = 256, Wd = 64, Hd = 64, N = 1024, G = 32;
  const int P = Wd * Hd;

  const float* graph  = (const float*)d_in[0];
  const float* image  = (const float*)d_in[1];
  const float* Wq     = (const float*)d_in[2];
  const float* bq     = (const float*)d_in[3];
  const float* Wk     = (const float*)d_in[4];
  const float* bk     = (const float*)d_in[5];
  const float* Wv     = (const float*)d_in[6];
  const float* bv     = (const float*)d_in[7];
  const float* c1w    = (const float*)d_in[8];
  const float* gamma  = (const float*)d_in[9];
  const float* beta   = (const float*)d_in[10];
  const float* mean   = (const float*)d_in[11];
  const float* var    = (const float*)d_in[12];
  const float* c2w    = (const float*)d_in[13];
  const float* c2b    = (const float*)d_in[14];
  const float* c3w    = (const float*)d_in[15];
  const float* c3b    = (const float*)d_in[16];
  float* out = (float*)d_out;

  // Workspace layout (floats). buf1 reused: Q -> msg -> h2 (disjoint lifetimes)
  float* ws = (float*)d_ws;
  const size_t szBCP = (size_t)B * C * P;
  const size_t szBCN = (size_t)B * C * N;
  const size_t szBPN = (size_t)B * P * N;
  float* buf1 = ws;
  float* bufH = buf1 + szBCP;
  float* Kbuf = bufH + szBCP;
  float* Vbuf = Kbuf + szBCN;
  float* Sbuf = Vbuf + szBCN;
  float* wtb  = Sbuf + szBPN;
  float* cmax = wtb + (size_t)9 * C * C;
  float* csum = cmax + (size_t)B * N;
  float* pmax = csum + (size_t)B * N;
  float* psum = pmax + (size_t)PCHUNK * B * N;

  dim3 blk(BD);
  const long gemmTiles = (long)B * (C / 32) * (P / 64);   // 4096
  const long attTiles  = (long)B * (P / 32) * (N / 64);   // 16384
  dim3 gGemm((unsigned)((gemmTiles + 7) / 8));
  dim3 gAtt((unsigned)((attTiles + 7) / 8));

  {  // transpose conv2 weights
    long tot = (long)C * C * 9;
    transpose_w9<<<dim3((unsigned)((tot + BD - 1) / BD)), blk, 0, stream>>>(c2w, wtb, C, C);
  }
  {  // K, V projections
    long tot = (long)B * C * N;
    kv_proj<<<dim3((unsigned)((tot + BD - 1) / BD)), blk, 0, stream>>>(
        graph, Wk, bk, Wv, bv, Kbuf, Vbuf, B, C, N, G);
  }
  // Q = (Wq x + bq)/16 -> buf1
  gemm_wx<0><<<gGemm, blk, 0, stream>>>(Wq, image, buf1, bq, nullptr, nullptr,
                                        nullptr, nullptr, B, C, C, P);
  // S = Q^T K
  att_gemm<<<gAtt, blk, 0, stream>>>(buf1, Kbuf, Sbuf, B, C, P, N);
  // softmax over P axis; fold 1/sum into V
  {
    dim3 gP((unsigned)(B * (N / 256) * PCHUNK));
    dim3 gR((unsigned)((B * N + BD - 1) / BD));
    sm_pmax<<<gP, blk, 0, stream>>>(Sbuf, pmax, B, P, N);
    sm_rmax<<<gR, blk, 0, stream>>>(pmax, cmax, B * N);
    sm_expsum<<<gP, blk, 0, stream>>>(Sbuf, cmax, psum, B, P, N);
    sm_rsum<<<gR, blk, 0, stream>>>(psum, csum, B * N);
    long tot = (long)B * C * N;
    v_scale<<<dim3((unsigned)((tot + BD - 1) / BD)), blk, 0, stream>>>(Vbuf, csum, B, C, N);
  }
  // msg = Vs * E^T -> buf1
  msg_gemm<<<gGemm, blk, 0, stream>>>(Vbuf, Sbuf, buf1, B, C, N, P);
  // h = LeakyReLU(BN(conv1(msg))) -> bufH
  gemm_wx<1><<<gGemm, blk, 0, stream>>>(c1w, buf1, bufH, gamma, beta, mean, var,
                                        nullptr, B, C, C, P);
  // h2 = conv2_3x3(h) + b2 -> buf1
  conv3x3_wmma<<<gGemm, blk, 0, stream>>>(wtb, bufH, c2b, buf1, B, C, C, Hd);
  // out = image + conv3(h2) + b3
  gemm_wx<3><<<gGemm, blk, 0, stream>>>(c3w, buf1, out, c3b, nullptr, nullptr,
                                        nullptr, image, B, C, C, P);
}